// OneUnkMaterialSparseStack_N_38448547234791
// MI455X (gfx1250) — compile-verified
//
#include <hip/hip_runtime.h>
#include <math.h>

// ---------------------------------------------------------------------------
// Workspace layout (float units). Instance sizes: R=350 -> NS=331, W=4000,
// A=64, L=22, M=6, D=500.  nmat table = 2*(M+1)*W floats = 224 KB total ws.
// ---------------------------------------------------------------------------
#define KS        20          // gaussian kernel taps
#define NK_OFF    0           // nk_s smoothed values (NS floats)
#define WMIN_OFF  336
#define WMAX_OFF  337
#define D_OFF     344         // layer thicknesses (up to 32)
#define NMAT_OFF  384         // (M+1) x W complex (float2), 8-byte aligned

// ---------------- complex helpers (fp32, matches jnp complex64) ------------
struct c2 { float x, y; };
__device__ __forceinline__ c2 cadd(c2 a, c2 b) { return {a.x + b.x, a.y + b.y}; }
__device__ __forceinline__ c2 csub(c2 a, c2 b) { return {a.x - b.x, a.y - b.y}; }
__device__ __forceinline__ c2 cmul(c2 a, c2 b) { return {a.x*b.x - a.y*b.y, a.x*b.y + a.y*b.x}; }
__device__ __forceinline__ c2 cscale(c2 a, float s) { return {a.x*s, a.y*s}; }
__device__ __forceinline__ c2 cdiv(c2 a, c2 b) {
    float inv = 1.0f / (b.x*b.x + b.y*b.y);
    return { (a.x*b.x + a.y*b.y)*inv, (a.y*b.x - a.x*b.y)*inv };
}
// principal-branch complex sqrt (imag sign follows sign of Im(z))
__device__ __forceinline__ c2 csqrtc(c2 z) {
    float mag = sqrtf(z.x*z.x + z.y*z.y);
    float re  = sqrtf(fmaxf(0.5f*(mag + z.x), 0.0f));
    float im  = sqrtf(fmaxf(0.5f*(mag - z.x), 0.0f));
    return { re, (z.y < 0.0f) ? -im : im };
}

#if defined(__HIP_DEVICE_COMPILE__) && defined(__gfx1250__) && \
    __has_builtin(__builtin_amdgcn_wmma_f32_16x16x4_f32)
#define USE_WMMA_CONV 1
typedef __attribute__((ext_vector_type(2))) float v2f;
typedef __attribute__((ext_vector_type(8))) float v8f;
#endif

// ---------------------------------------------------------------------------
// Kernel 1 (one wave32): wavelength min/max reduction, thickness vector,
// Gaussian taps, and the 20-tap "valid" convolution done as a tiled GEMM on
// V_WMMA_F32_16X16X4_F32.  B broadcasts taps across all 16 columns, so every
// column of C equals conv output (robust to B's N-lane mapping).
// ---------------------------------------------------------------------------
__global__ __launch_bounds__(32) void prep_kernel(
    const float* __restrict__ refr, const float* __restrict__ comp,
    const float* __restrict__ above, const float* __restrict__ below,
    const float* __restrict__ wl, const float* __restrict__ unk_th,
    float* __restrict__ ws, int R, int W, int nAbove, int nBelow)
{
    int lane = threadIdx.x;

    // --- min/max of wavelengths (wave32 shuffle reduce) ---
    float lmin = 3.4e38f, lmax = -3.4e38f;
    for (int i = lane; i < W; i += 32) {
        float v = wl[i];
        lmin = fminf(lmin, v); lmax = fmaxf(lmax, v);
    }
    for (int off = 16; off > 0; off >>= 1) {
        lmin = fminf(lmin, __shfl_xor(lmin, off, 32));
        lmax = fmaxf(lmax, __shfl_xor(lmax, off, 32));
    }
    if (lane == 0) { ws[WMIN_OFF] = lmin; ws[WMAX_OFF] = lmax; }

    // --- thickness vector d = [above, unk/1000*comp, below] ---
    int nd = nAbove + 1 + nBelow;
    if (lane < nd) {
        float v;
        if (lane < nAbove)       v = above[lane];
        else if (lane == nAbove) v = unk_th[0] * (1.0f/1000.0f) * comp[0];
        else                     v = below[lane - nAbove - 1];
        ws[D_OFF + lane] = v;
    }

    // --- reversed, normalized gaussian taps h[j] = g[KS-1-j]/sum(g) ---
    __shared__ float h[32];
    float gv = 0.0f;
    if (lane < KS) {
        float xk = -10.0f + 20.0f * (float)lane / (float)(KS - 1);
        float xs = xk * 0.25f;                  // x / SIGMA
        gv = 0.5f * expf(-0.5f * xs * xs);      // AMPLITUDE * exp(...)
    }
    float gs = gv;
    for (int off = 16; off > 0; off >>= 1) gs += __shfl_xor(gs, off, 32);
    if (lane < KS) h[(KS - 1) - lane] = gv / gs;
    __syncthreads();

    int NS = R - KS + 1;                        // 331
    int hi2 = lane >> 4;                        // lane half (0/1)
    int l16 = lane & 15;                        // row within tile
    int ntiles = (NS + 15) / 16;

#ifdef USE_WMMA_CONV
    // A 16x4 f32 layout: lanes 0-15 hold M=0..15 with K={v, } and lanes 16-31
    // hold K={2+v}: element index K = 4*kk + 2*hi2 + component.
    for (int mt = 0; mt < ntiles; ++mt) {
        int base = mt * 16 + l16;
        v8f c = {0.f, 0.f, 0.f, 0.f, 0.f, 0.f, 0.f, 0.f};
        for (int kk = 0; kk < KS / 4; ++kk) {
            int k0 = kk * 4 + 2 * hi2;
            int ia = base + k0;
            int ia0 = min(ia, R - 1);
            int ia1 = min(ia + 1, R - 1);
            v2f av, bv;
            float x0 = refr[ia0], x1 = refr[ia1];
            av.x = (ia     < R) ? x0 : 0.0f;
            av.y = (ia + 1 < R) ? x1 : 0.0f;
            bv.x = h[k0];                       // taps broadcast over N
            bv.y = h[k0 + 1];
            c = __builtin_amdgcn_wmma_f32_16x16x4_f32(
                    false, av, false, bv, (short)0, c, false, false);
        }
        // C/D: lane L, vgpr j -> row = j + 8*(L/16), col = L%16. Col 0 writes.
        if (l16 == 0) {
            #pragma unroll
            for (int j = 0; j < 8; ++j) {
                int row = mt * 16 + j + 8 * hi2;
                if (row < NS) ws[NK_OFF + row] = c[j];
            }
        }
    }
#else
    for (int i = lane; i < NS; i += 32) {
        float s = 0.0f;
        for (int j = 0; j < KS; ++j) s += refr[i + j] * h[j];
        ws[NK_OFF + i] = s;
    }
#endif
}

// ---------------------------------------------------------------------------
// Kernel 2: per-wavelength material table -> nmat[(M+1) x W] complex.
// Materials 0..M-1: binary-search interp of (grid, n, k).  Material M: the
// smoothed unknown interpolated on linspace(wlmin, wlmax, NS) knots.
// ---------------------------------------------------------------------------
__global__ __launch_bounds__(256) void nmat_kernel(
    const float* __restrict__ fx, const float* __restrict__ wl,
    float* __restrict__ ws, int W, int M, int D, int NS)
{
    int w = blockIdx.x * blockDim.x + threadIdx.x;
    if (w >= W) return;
    float x = wl[w];
    float2* nmat = (float2*)(ws + NMAT_OFF);

    for (int m = 0; m < M; ++m) {
        const float* xp = fx + (size_t)(m * 3 + 0) * D;
        const float* fn = fx + (size_t)(m * 3 + 1) * D;
        const float* fk = fx + (size_t)(m * 3 + 2) * D;
        int lo = 0, hi = D - 1;
        while (hi - lo > 1) {
            int mid = (lo + hi) >> 1;
            if (xp[mid] <= x) lo = mid; else hi = mid;
        }
        float x0 = xp[lo], x1 = xp[lo + 1];
        float t = (x - x0) / (x1 - x0);
        t = fminf(fmaxf(t, 0.0f), 1.0f);        // jnp.interp clamp semantics
        float nre = fn[lo] + t * (fn[lo + 1] - fn[lo]);
        float nim = fk[lo] + t * (fk[lo + 1] - fk[lo]);
        nmat[(size_t)m * W + w] = make_float2(nre, nim);
    }

    float wmin = ws[WMIN_OFF], wmax = ws[WMAX_OFF];
    float xx  = fminf(fmaxf(x, wmin), wmax);
    float pos = (xx - wmin) * (float)(NS - 1) / (wmax - wmin);
    int i = (int)pos;
    i = max(0, min(i, NS - 2));
    float fr = pos - (float)i;
    float v = ws[NK_OFF + i] + fr * (ws[NK_OFF + i + 1] - ws[NK_OFF + i]);
    nmat[(size_t)M * W + w] = make_float2(v, 0.0f);
}

// ---------------------------------------------------------------------------
// Kernel 3: one thread per (angle, wavelength); sequential 2x2 complex chain.
// w fastest -> all nmat/wl/out accesses fully coalesced per wave32.
// ---------------------------------------------------------------------------
__global__ __launch_bounds__(256) void tmm_kernel(
    const float* __restrict__ wl, const float* __restrict__ aoi,
    const int* __restrict__ md, const int* __restrict__ polp,
    const float* __restrict__ ws, float* __restrict__ out,
    int W, int A, int L)
{
    int tid = blockIdx.x * blockDim.x + threadIdx.x;
    int total = A * W;
    if (tid >= total) return;
    int a = tid / W, w = tid - a * W;
    const float2* nmat = (const float2*)(ws + NMAT_OFF);
    bool pp = (polp[0] != 0);

    float wlv = wl[w];
    float kf  = 6.28318530717958648f / wlv;     // 2*pi / lambda
    float th0 = aoi[a] * (3.14159265358979324f / 180.0f);
    float s0  = sinf(th0);

    // ambient layer (index 0)
    float2 tv0 = nmat[(size_t)md[0] * W + w];
    c2 n_p = {tv0.x, tv0.y};
    c2 sin0 = cscale(n_p, s0);                  // n0 * sin(theta0)
    c2 st  = cdiv(sin0, n_p);
    c2 om  = {1.0f - (st.x*st.x - st.y*st.y), -2.0f*st.x*st.y};
    c2 c_p = csqrtc(om);
    c2 f_p = cmul(n_p, c_p);
    c2 k_p = cscale(f_p, kf);
    float denom0 = pp ? (n_p.x*c_p.x + n_p.y*c_p.y) : f_p.x;  // Re(n*conj(c)) / Re(n*c)

    __builtin_prefetch(&nmat[(size_t)md[1] * W + w], 0, 3);   // global_prefetch_b8

    c2 m00, m01, m10, m11;
    for (int l = 1; l < L; ++l) {
        float2 tv = nmat[(size_t)md[l] * W + w];
        c2 n_c = {tv.x, tv.y};
        c2 st2 = cdiv(sin0, n_c);
        c2 om2 = {1.0f - (st2.x*st2.x - st2.y*st2.y), -2.0f*st2.x*st2.y};
        c2 c_c = csqrtc(om2);
        c2 f_c = cmul(n_c, c_c);
        c2 k_c = cscale(f_c, kf);

        c2 sum, diff;
        if (!pp) {                              // s-polarization
            sum = cadd(f_p, f_c); diff = csub(f_p, f_c);
        } else {                                // p-polarization
            c2 num = cmul(n_c, c_p);
            c2 den = cmul(n_p, c_c);
            sum = cadd(num, den); diff = csub(num, den);
        }
        c2 r = cdiv(diff, sum);
        c2 t = cdiv(cscale(f_p, 2.0f), sum);    // 2*n_prev*cos_prev / sum

        if (l == 1) {
            m00 = cdiv((c2){1.0f, 0.0f}, t);
            m01 = cdiv(r, t);
            m10 = m01; m11 = m00;
        } else {
            float dl = ws[D_OFF + (l - 2)];
            c2 delta = cscale(k_p, dl);         // kz of previous (interior) layer
            float sx, cx;
            sincosf(delta.x, &sx, &cx);
            float eyp = expf(delta.y);
            float eyn = expf(-delta.y);
            c2 em = cdiv((c2){eyp * cx, -eyp * sx}, t);  // exp(-i*delta)/t
            c2 ep = cdiv((c2){eyn * cx,  eyn * sx}, t);  // exp(+i*delta)/t
            c2 bq = cmul(r, em);
            c2 cq = cmul(r, ep);
            c2 nm00 = cadd(cmul(m00, em), cmul(m01, cq));
            c2 nm01 = cadd(cmul(m00, bq), cmul(m01, ep));
            c2 nm10 = cadd(cmul(m10, em), cmul(m11, cq));
            c2 nm11 = cadd(cmul(m10, bq), cmul(m11, ep));
            m00 = nm00; m01 = nm01; m10 = nm10; m11 = nm11;
        }
        n_p = n_c; c_p = c_c; f_p = f_c; k_p = k_c;
    }

    c2 rr = cdiv(m10, m00);
    c2 tt = cdiv((c2){1.0f, 0.0f}, m00);
    float R = rr.x*rr.x + rr.y*rr.y;
    float denomL = pp ? (n_p.x*c_p.x + n_p.y*c_p.y) : f_p.x;
    float T = (tt.x*tt.x + tt.y*tt.y) * denomL / denom0;

    size_t AW = (size_t)A * W;
    out[tid]          = R;
    out[AW + tid]     = T;
    out[2 * AW + tid] = 1.0f - R - T;
}

// ---------------------------------------------------------------------------
extern "C" void kernel_launch(void* const* d_in, const int* in_sizes, int n_in,
                              void* d_out, int out_size, void* d_ws, size_t ws_size,
                              hipStream_t stream) {
    const float* refr = (const float*)d_in[0];
    const float* comp = (const float*)d_in[1];
    const float* fx   = (const float*)d_in[2];
    const float* abv  = (const float*)d_in[3];
    const float* blw  = (const float*)d_in[4];
    const float* wl   = (const float*)d_in[5];
    const float* aoi  = (const float*)d_in[6];
    const int*   md   = (const int*)d_in[7];
    const int*   pol  = (const int*)d_in[8];
    const float* uth  = (const float*)d_in[9];

    int R   = in_sizes[0];          // 350
    int W   = in_sizes[5];          // 4000
    int A   = in_sizes[6];          // 64
    int L   = in_sizes[7];          // 22 layers
    int nAb = in_sizes[3];          // 10
    int nBl = in_sizes[4];          // 9
    const int M = 6;
    int D   = in_sizes[2] / (3 * M);// 500
    int NS  = R - KS + 1;           // 331

    float* ws = (float*)d_ws;

    prep_kernel<<<1, 32, 0, stream>>>(refr, comp, abv, blw, wl, uth,
                                      ws, R, W, nAb, nBl);
    nmat_kernel<<<(W + 255) / 256, 256, 0, stream>>>(fx, wl, ws, W, M, D, NS);
    int total = A * W;
    tmm_kernel<<<(total + 255) / 256, 256, 0, stream>>>(
        wl, aoi, md, pol, ws, (float*)d_out, W, A, L);
}